// SRNNGRUCell_82978768158899
// MI455X (gfx1250) — compile-verified
//
#include <hip/hip_runtime.h>
#include <math.h>

typedef __attribute__((ext_vector_type(2))) float v2f;
typedef __attribute__((ext_vector_type(8))) float v8f;
typedef __attribute__((ext_vector_type(4))) int   v4i;

#define H           4096
#define THREEH      12288
#define KTILE       64
#define ROWS_PER_WG 128
#define LDS_STRIDE  68        // 64 + 4 pad: float4-aligned rows, conflict-free b64 reads
#define KSPLIT      2
#define KSEG        (H / KSPLIT)     // 2048
#define NTILES      (KSEG / KTILE)   // 32
#define BLKS_PER_MAT 96              // 12288 / 128

// ---- CDNA5 async global->LDS path (ASYNCcnt), with portable fallback -------
#ifdef __has_builtin
#if __has_builtin(__builtin_amdgcn_global_load_async_to_lds_b128)
#define HAVE_ASYNC_LDS 1
#endif
#endif
#ifndef HAVE_ASYNC_LDS
#define HAVE_ASYNC_LDS 0
#endif

__device__ __forceinline__ void async_cp16(const float* g, float* l) {
#if HAVE_ASYNC_LDS
    __builtin_amdgcn_global_load_async_to_lds_b128(
        (__attribute__((address_space(1))) v4i*)g,
        (__attribute__((address_space(3))) v4i*)l, 0, 0);
#else
    (void)g; (void)l;
#endif
}

__device__ __forceinline__ void wait_async_le8() {
#if HAVE_ASYNC_LDS
#if __has_builtin(__builtin_amdgcn_s_wait_asynccnt)
    __builtin_amdgcn_s_wait_asynccnt(8);
#else
    asm volatile("s_wait_asynccnt 0x8" ::: "memory");
#endif
#endif
}
__device__ __forceinline__ void wait_async_le0() {
#if HAVE_ASYNC_LDS
#if __has_builtin(__builtin_amdgcn_s_wait_asynccnt)
    __builtin_amdgcn_s_wait_asynccnt(0);
#else
    asm volatile("s_wait_asynccnt 0x0" ::: "memory");
#endif
#endif
}

// ---------------------------------------------------------------------------
// Stage A: bandwidth-bound GEMV partials via V_WMMA_F32_16X16X4_F32.
// 384 WGs: {ksplit half} x {w_ih | w_hh} x {96 row-blocks of 128 rows}.
// Double-buffered LDS tiles; next tile streamed with async-to-LDS while
// WMMA consumes the current one.  A = 16x4 weight fragment, B = vector
// chunk broadcast; column N=0 of the f32 accumulator = 16 dot products.
// ---------------------------------------------------------------------------
__global__ __launch_bounds__(256)
void gemv_wmma_kernel(const float* __restrict__ w_ih, const float* __restrict__ w_hh,
                      const float* __restrict__ x,    const float* __restrict__ hid,
                      float* __restrict__ part)
{
    __shared__ __attribute__((aligned(16))) float tile[2][ROWS_PER_WG * LDS_STRIDE];
    __shared__ __attribute__((aligned(16))) float vlds[2][KTILE];

    const int tid = threadIdx.x;
    const int b   = (int)blockIdx.x;
    const int ks  = b / (2 * BLKS_PER_MAT);          // 0..1  (K half)
    const int r   = b % (2 * BLKS_PER_MAT);
    const bool is_hh = r >= BLKS_PER_MAT;
    const float* __restrict__ W = is_hh ? w_hh : w_ih;
    const float* __restrict__ v = is_hh ? hid  : x;
    const int rowBase = (is_hh ? r - BLKS_PER_MAT : r) * ROWS_PER_WG;
    const int kbase   = ks * KSEG;
    float* __restrict__ outp = part + (size_t)(ks * 2 + (is_hh ? 1 : 0)) * THREEH;

    const int wave   = tid >> 5;
    const int lane   = tid & 31;
    const int lane16 = lane & 15;
    const int koff   = (lane >> 4) << 1;   // 0: K=0,1   2: K=2,3

    v8f acc = {};
    int cur = 0;

#if HAVE_ASYNC_LDS
    auto issue = [&](int k0, float* buf) {
#pragma unroll
        for (int i = 0; i < 8; ++i) {
            int f   = i * 256 + tid;
            int row = f >> 4;
            int c4  = (f & 15) << 2;
            async_cp16(W + (size_t)(rowBase + row) * H + k0 + c4,
                       buf + row * LDS_STRIDE + c4);
        }
    };
    issue(kbase, tile[0]);
    if (tid < KTILE) vlds[0][tid] = v[kbase + tid];
    for (int it = 0; it < NTILES; ++it) {
        const int k0 = kbase + it * KTILE;
        __syncthreads();                       // nobody still reading tile[cur^1]
        if (it + 1 < NTILES) {
            issue(k0 + KTILE, tile[cur ^ 1]);  // stream next tile during compute
            if (tid < KTILE) vlds[cur ^ 1][tid] = v[k0 + KTILE + tid];
            wait_async_le8();                  // tile 'it' (this wave) retired
        } else {
            wait_async_le0();
        }
        __syncthreads();                       // tile 'it' visible to all waves
        const float* arow = tile[cur] + (wave * 16 + lane16) * LDS_STRIDE + koff;
        const float* vrow = vlds[cur] + koff;
#pragma unroll
        for (int kk = 0; kk < KTILE; kk += 4) {
            v2f af = *(const v2f*)(arow + kk);
            v2f bf = *(const v2f*)(vrow + kk);
            acc = __builtin_amdgcn_wmma_f32_16x16x4_f32(
                      false, af, false, bf, (short)0, acc, false, false);
        }
        cur ^= 1;
    }
#else
    // Register-staged double buffer (same pipeline, no async path available)
    float4 regs[8];
    auto gload = [&](int k0) {
#pragma unroll
        for (int i = 0; i < 8; ++i) {
            int f = i * 256 + tid;
            regs[i] = *(const float4*)(W + (size_t)(rowBase + (f >> 4)) * H + k0 + ((f & 15) << 2));
        }
    };
    auto commit = [&](float* buf) {
#pragma unroll
        for (int i = 0; i < 8; ++i) {
            int f = i * 256 + tid;
            *(float4*)(buf + (f >> 4) * LDS_STRIDE + ((f & 15) << 2)) = regs[i];
        }
    };
    gload(kbase);
    for (int it = 0; it < NTILES; ++it) {
        const int k0 = kbase + it * KTILE;
        __syncthreads();
        commit(tile[cur]);
        if (tid < KTILE) vlds[cur][tid] = v[k0 + tid];
        if (it + 1 < NTILES) gload(k0 + KTILE);   // overlap with compute below
        __syncthreads();
        const float* arow = tile[cur] + (wave * 16 + lane16) * LDS_STRIDE + koff;
        const float* vrow = vlds[cur] + koff;
#pragma unroll
        for (int kk = 0; kk < KTILE; kk += 4) {
            v2f af = *(const v2f*)(arow + kk);
            v2f bf = *(const v2f*)(vrow + kk);
            acc = __builtin_amdgcn_wmma_f32_16x16x4_f32(
                      false, af, false, bf, (short)0, acc, false, false);
        }
        cur ^= 1;
    }
#endif

    // C/D layout: VGPR p, lanes 0-15: M=p, N=lane; lanes 16-31: M=p+8.
    if (lane16 == 0) {
        const int mbase = wave * 16 + ((lane >> 4) << 3);
#pragma unroll
        for (int p = 0; p < 8; ++p)
            outp[rowBase + mbase + p] = acc[p];
    }
}

// ---------------------------------------------------------------------------
// Stage B: merge K-split partials + bias, gates, relu, inclusive scan.
// ---------------------------------------------------------------------------
__device__ __forceinline__ float sigmoidf_(float v) { return 1.0f / (1.0f + expf(-v)); }

__global__ __launch_bounds__(1024)
void gates_scan_kernel(const float* __restrict__ part,
                       const float* __restrict__ b_ih, const float* __restrict__ b_hh,
                       const float* __restrict__ hid,  float* __restrict__ S)
{
    __shared__ float psum[1024];
    const float* gi0 = part;                  // ks0, w_ih
    const float* gh0 = part + THREEH;         // ks0, w_hh
    const float* gi1 = part + 2 * THREEH;     // ks1, w_ih
    const float* gh1 = part + 3 * THREEH;     // ks1, w_hh

    const int tid = threadIdx.x;
    const int j0  = tid * 4;
    float a[4];
#pragma unroll
    for (int e = 0; e < 4; ++e) {
        int j  = j0 + e;
        float ir = gi0[j]         + gi1[j]         + b_ih[j];
        float iz = gi0[H + j]     + gi1[H + j]     + b_ih[H + j];
        float in = gi0[2 * H + j] + gi1[2 * H + j] + b_ih[2 * H + j];
        float hr = gh0[j]         + gh1[j]         + b_hh[j];
        float hz = gh0[H + j]     + gh1[H + j]     + b_hh[H + j];
        float hn = gh0[2 * H + j] + gh1[2 * H + j] + b_hh[2 * H + j];
        float rr = sigmoidf_(ir + hr);
        float zz = sigmoidf_(iz + hz);
        float nn = tanhf(in + rr * hn);
        float bb = (1.0f - zz) * hid[j] + zz * nn;
        a[e] = fmaxf(bb, 0.0f);
    }
    a[1] += a[0]; a[2] += a[1]; a[3] += a[2];
    psum[tid] = a[3];
    __syncthreads();
    for (int off = 1; off < 1024; off <<= 1) {
        float add = (tid >= off) ? psum[tid - off] : 0.0f;
        __syncthreads();
        psum[tid] += add;
        __syncthreads();
    }
    const float excl = psum[tid] - a[3];
#pragma unroll
    for (int e = 0; e < 4; ++e)
        S[j0 + e] = excl + a[e];
}

// ---------------------------------------------------------------------------
// Stage C: outputs[t][j] = S[t] + hidden[(j - t - 1) mod 4096].
// Row 4095 is also new_hidden.
// ---------------------------------------------------------------------------
__global__ __launch_bounds__(256)
void outputs_kernel(const float* __restrict__ S, const float* __restrict__ hid,
                    float* __restrict__ out)
{
    const int   t = (int)blockIdx.x;
    const float s = S[t];
    float* __restrict__ row = out + (size_t)t * H;
    const int shift = t + 1;
#pragma unroll
    for (int it = 0; it < 4; ++it) {
        int j = ((it << 8) + (int)threadIdx.x) << 2;
        float4 val;
        val.x = s + hid[(unsigned)(j + 0 - shift) & (H - 1)];
        val.y = s + hid[(unsigned)(j + 1 - shift) & (H - 1)];
        val.z = s + hid[(unsigned)(j + 2 - shift) & (H - 1)];
        val.w = s + hid[(unsigned)(j + 3 - shift) & (H - 1)];
        *(float4*)(row + j) = val;
        if (t == H - 1)
            *(float4*)(out + (size_t)H * H + j) = val;
    }
}

// ---------------------------------------------------------------------------
extern "C" void kernel_launch(void* const* d_in, const int* in_sizes, int n_in,
                              void* d_out, int out_size, void* d_ws, size_t ws_size,
                              hipStream_t stream)
{
    const float* x    = (const float*)d_in[0];
    const float* hid  = (const float*)d_in[1];
    const float* w_ih = (const float*)d_in[2];
    const float* w_hh = (const float*)d_in[3];
    const float* b_ih = (const float*)d_in[4];
    const float* b_hh = (const float*)d_in[5];
    float* out = (float*)d_out;

    float* part = (float*)d_ws;                 // 4 * 12288 floats of GEMV partials
    float* S    = part + 4 * THREEH;            // 4096 floats (~208 KB total ws)

    gemv_wmma_kernel<<<KSPLIT * 2 * BLKS_PER_MAT, 256, 0, stream>>>(
        w_ih, w_hh, x, hid, part);
    gates_scan_kernel<<<1, 1024, 0, stream>>>(part, b_ih, b_hh, hid, S);
    outputs_kernel<<<H, 256, 0, stream>>>(S, hid, out);
}